// GraphAttention_v2_30167850287778
// MI455X (gfx1250) — compile-verified
//
#include <hip/hip_runtime.h>
#include <hip/hip_bf16.h>
#include <math.h>

// Problem constants (match reference): N=2048, F=UNITS=128, FE=16, DEG=64.
#define NN    2048
#define FD    128
#define FE_D  16
#define DEG   64

typedef __attribute__((ext_vector_type(2))) float v2f;
typedef __attribute__((ext_vector_type(8))) float v8f;

// ---------------------------------------------------------------------------
// Kernel 1: hl = ns @ W_left, hr = ns @ W_right via V_WMMA_F32_16X16X4_F32.
// One wave (32 threads) per 16x16 output tile; the A fragment (node_states)
// is shared between the two WMMAs (same tile of hl and hr).
//
// Register layouts per CDNA5 ISA 7.12.2:
//   A (16x4 f32): lane<16 -> M=lane, VGPR0=K+0, VGPR1=K+1
//                 lane>=16 -> M=lane-16, VGPR0=K+2, VGPR1=K+3
//   B (4x16 f32): mirrored, N=lane%16, K split by lane-half the same way.
//   C/D (16x16 f32): VGPR v -> M = v + 8*(lane>=16), N = lane%16.
// ---------------------------------------------------------------------------
__global__ void __launch_bounds__(32)
gat_gemm_wmma(const float* __restrict__ ns,
              const float* __restrict__ Wl,
              const float* __restrict__ Wr,
              float* __restrict__ hl,
              float* __restrict__ hr) {
    const int lane  = threadIdx.x & 31;
    const int half  = lane >> 4;      // 0: lanes 0-15, 1: lanes 16-31
    const int l16   = lane & 15;
    const int tileN = blockIdx.x & 7;             // 128/16 = 8 N-tiles
    const int tileM = blockIdx.x >> 3;            // 2048/16 = 128 M-tiles
    const int m     = tileM * 16 + l16;           // A row for this lane
    const int n     = tileN * 16 + l16;           // B/C column for this lane

    v8f cl = {};
    v8f cr = {};

    #pragma unroll 4
    for (int k = 0; k < FD; k += 4) {
        const int ka = k + half * 2;
        // A fragment: two consecutive K elements -> single b64 load.
        v2f a = *(const v2f*)(ns + (size_t)m * FD + ka);
        // B fragments: W[k][n] with row stride FD (two b32 loads each).
        v2f bl, br;
        bl.x = Wl[(size_t)(ka + 0) * FD + n];
        bl.y = Wl[(size_t)(ka + 1) * FD + n];
        br.x = Wr[(size_t)(ka + 0) * FD + n];
        br.y = Wr[(size_t)(ka + 1) * FD + n];
        cl = __builtin_amdgcn_wmma_f32_16x16x4_f32(false, a, false, bl,
                                                   (short)0, cl, false, false);
        cr = __builtin_amdgcn_wmma_f32_16x16x4_f32(false, a, false, br,
                                                   (short)0, cr, false, false);
    }

    #pragma unroll
    for (int v = 0; v < 8; ++v) {
        const int row = tileM * 16 + v + half * 8;
        hl[(size_t)row * FD + n] = cl[v];
        hr[(size_t)row * FD + n] = cr[v];
    }
}

// ---------------------------------------------------------------------------
// Kernel 2: per-node attention partials.
//   sL[i] = sum_f W_attn[f]     * lrelu(hl[i][f])
//   sR[i] = sum_f W_attn[128+f] * lrelu(hr[i][f])
//   a[k]  = sum_c W_edge[k][c]  * W_attn[256+c]   (16-vector, folds the
//           entire 131072x16x16 edge GEMM out of existence)
// One wave32 per node, 8 nodes per 256-thread block.
// ---------------------------------------------------------------------------
__global__ void __launch_bounds__(256)
gat_node_prep(const float* __restrict__ hl,
              const float* __restrict__ hr,
              const float* __restrict__ W_attn,
              const float* __restrict__ W_edge,
              float* __restrict__ sL,
              float* __restrict__ sR,
              float* __restrict__ a_vec) {
    const int wave = threadIdx.x >> 5;
    const int lane = threadIdx.x & 31;
    const int node = blockIdx.x * 8 + wave;

    float accL = 0.f, accR = 0.f;
    #pragma unroll
    for (int t = 0; t < 4; ++t) {
        const int f = lane + t * 32;
        float xl = hl[(size_t)node * FD + f];
        float xr = hr[(size_t)node * FD + f];
        xl = (xl > 0.f) ? xl : 0.2f * xl;   // leaky_relu(0.2)
        xr = (xr > 0.f) ? xr : 0.2f * xr;
        accL += xl * W_attn[f];
        accR += xr * W_attn[FD + f];
    }
    #pragma unroll
    for (int off = 16; off > 0; off >>= 1) {
        accL += __shfl_xor(accL, off, 32);
        accR += __shfl_xor(accR, off, 32);
    }
    if (lane == 0) {
        sL[node] = accL;
        sR[node] = accR;
    }

    if (blockIdx.x == 0 && threadIdx.x < FE_D) {
        float s = 0.f;
        #pragma unroll
        for (int c = 0; c < FE_D; ++c)
            s += W_edge[threadIdx.x * FE_D + c] * W_attn[2 * FD + c];
        a_vec[threadIdx.x] = s;
    }
}

// ---------------------------------------------------------------------------
// Kernel 3: per-node edge scores + weighted aggregation.
// One 128-thread block per source node i (its 64 edges are contiguous).
//   s_j   = exp(clip(sL[i] + sR[d_j] + ef[i,d_j]·a, -2, 2))
//   out[i]= (sum_j s_j * hr[d_j]) / sum_j s_j
// hr is 1 MB -> fully L2-resident; ef gathers are the only HBM-side traffic.
// ---------------------------------------------------------------------------
__global__ void __launch_bounds__(128)
gat_edge_aggregate(const float* __restrict__ hr,
                   const float* __restrict__ ef,
                   const int*   __restrict__ edges,
                   const float* __restrict__ sL,
                   const float* __restrict__ sR,
                   const float* __restrict__ a_vec,
                   float* __restrict__ out) {
    __shared__ int   s_dst[DEG];
    __shared__ float s_sc[DEG];
    __shared__ float s_a[FE_D];
    __shared__ float s_inv;

    const int i = blockIdx.x;
    const int t = threadIdx.x;

    if (t < FE_D) s_a[t] = a_vec[t];
    if (t < DEG)  s_dst[t] = edges[((size_t)i * DEG + t) * 2 + 1];
    __syncthreads();

    if (t < DEG) {
        const int d = s_dst[t];
        const float4* p = (const float4*)(ef + ((size_t)i * NN + d) * FE_D);
        float dot = 0.f;
        #pragma unroll
        for (int q = 0; q < 4; ++q) {
            float4 v = p[q];
            dot += v.x * s_a[4 * q + 0] + v.y * s_a[4 * q + 1] +
                   v.z * s_a[4 * q + 2] + v.w * s_a[4 * q + 3];
        }
        float sc = sL[i] + sR[d] + dot;
        sc = fminf(2.0f, fmaxf(-2.0f, sc));
        s_sc[t] = __expf(sc);
    }
    __syncthreads();

    if (t == 0) {
        float s = 0.f;
        #pragma unroll
        for (int j = 0; j < DEG; ++j) s += s_sc[j];
        s_inv = 1.0f / s;          // s >= 64*e^-2 > 0, always safe
    }
    __syncthreads();

    // Thread t owns output feature t: coalesced 512B row reads from L2.
    float acc = 0.f;
    #pragma unroll 4
    for (int j = 0; j < DEG; ++j) {
        if (j + 4 < DEG)
            __builtin_prefetch(hr + (size_t)s_dst[j + 4] * FD + t, 0, 0);
        acc += s_sc[j] * hr[(size_t)s_dst[j] * FD + t];
    }
    out[(size_t)i * FD + t] = acc * s_inv;
}

// ---------------------------------------------------------------------------
// Launch wrapper. Inputs (setup_inputs order):
//   0 node_states (1,N,F) f32   1 edges (1,E,2) i32   2 edge_features (1,N,N,FE) f32
//   3 W_left (F,U) f32          4 W_right (F,U) f32   5 W_attn (2U+FE,1) f32
//   6 W_edge (FE,FE) f32
// Output: (1,N,U) f32.  Workspace: hl | hr | sL | sR | a_vec.
// ---------------------------------------------------------------------------
extern "C" void kernel_launch(void* const* d_in, const int* in_sizes, int n_in,
                              void* d_out, int out_size, void* d_ws, size_t ws_size,
                              hipStream_t stream) {
    const float* ns     = (const float*)d_in[0];
    const int*   edges  = (const int*)  d_in[1];
    const float* ef     = (const float*)d_in[2];
    const float* Wl     = (const float*)d_in[3];
    const float* Wr     = (const float*)d_in[4];
    const float* W_attn = (const float*)d_in[5];
    const float* W_edge = (const float*)d_in[6];
    float* out = (float*)d_out;

    float* ws    = (float*)d_ws;
    float* hl    = ws;                         // N*F
    float* hr    = hl + (size_t)NN * FD;       // N*F
    float* sL    = hr + (size_t)NN * FD;       // N
    float* sR    = sL + NN;                    // N
    float* a_vec = sR + NN;                    // FE

    // 1) dense projections via f32 WMMA: 128 M-tiles x 8 N-tiles, 1 wave each
    gat_gemm_wmma<<<dim3((NN / 16) * (FD / 16)), dim3(32), 0, stream>>>(
        ns, Wl, Wr, hl, hr);

    // 2) per-node attention partials + folded edge-attention vector
    gat_node_prep<<<dim3(NN / 8), dim3(256), 0, stream>>>(
        hl, hr, W_attn, W_edge, sL, sR, a_vec);

    // 3) edge scores + normalized aggregation, one block per source node
    gat_edge_aggregate<<<dim3(NN), dim3(128), 0, stream>>>(
        hr, ef, edges, sL, sR, a_vec, out);
}